// ECALayer_49709951484475
// MI455X (gfx1250) — compile-verified
//
#include <hip/hip_runtime.h>
#include <math.h>

// Problem constants (from reference)
#define NB      8        // batch
#define CM      512      // D_MODEL channels
#define HW      16384    // 128*128
#define DI      1024     // D_INNER
#define DTR     32       // DT_RANK
#define DST     16       // D_STATE

typedef __attribute__((ext_vector_type(2))) float v2f;
typedef __attribute__((ext_vector_type(8))) float v8f;

// ---------------------------------------------------------------------------
// Kernel 1: global average pool. Grid = 16*512 blocks (rows 8..15 zero-pad the
// WMMA A-matrix), 256 threads, float4 coalesced loads. Reads 256 MB.
// pooled layout: [16][512] row-major (M x K for GEMM1).
// ---------------------------------------------------------------------------
__global__ void pool_mean_kernel(const float* __restrict__ x,
                                 float* __restrict__ pooled) {
    int bc = blockIdx.x;            // b = bc>>9, c = bc&511
    int t  = threadIdx.x;
    if ((bc >> 9) >= NB) {          // zero padding rows 8..15
        if (t == 0) pooled[bc] = 0.0f;
        return;
    }
    const float4* xr = (const float4*)(x + (size_t)bc * HW);  // 4096 float4
    float s = 0.0f;
#pragma unroll
    for (int k = 0; k < 16; ++k) {
        float4 v = xr[t + k * 256];
        s += v.x + v.y + v.z + v.w;
    }
    __shared__ float red[256];
    red[t] = s;
    __syncthreads();
    for (int off = 128; off > 0; off >>= 1) {
        if (t < off) red[t] += red[t + off];
        __syncthreads();
    }
    if (t == 0) pooled[bc] = red[0] * (1.0f / 16384.0f);
}

// ---------------------------------------------------------------------------
// Kernel 2: xz = pooled(16x512) @ W_in^T(512x2048) via V_WMMA_F32_16X16X4_F32.
// One wave per 16-column tile; 128 tiles; 4 waves/block -> 32 blocks x 128 thr.
// A[m][k] = pooled[m*512+k]; B[k][n] = W_in[(n0+n)*512 + k].
// f32 A 16x4: lane l -> row (l&15), K = 2*(l>>4)+{0,1}. B 4x16 mirrors it.
// Only D rows 0..7 (lanes 0..15, VGPRs 0..7) are real batches -> stored.
// ---------------------------------------------------------------------------
__global__ void gemm_xz_wmma(const float* __restrict__ pooled,
                             const float* __restrict__ W_in,
                             float* __restrict__ xz /*16 x 2048*/) {
    int lane = threadIdx.x & 31;
    int wave = threadIdx.x >> 5;
    int n0   = (blockIdx.x * 4 + wave) * 16;
    int r    = lane & 15;
    int kh   = (lane >> 4) * 2;
    const float* arow = pooled + r * CM + kh;
    const float* brow = W_in + (size_t)(n0 + r) * CM + kh;
    v8f acc = {0.f, 0.f, 0.f, 0.f, 0.f, 0.f, 0.f, 0.f};
#pragma unroll 4
    for (int k0 = 0; k0 < CM; k0 += 4) {
        v2f a = *(const v2f*)(arow + k0);
        v2f b = *(const v2f*)(brow + k0);
        acc = __builtin_amdgcn_wmma_f32_16x16x4_f32(
            /*neg_a=*/false, a, /*neg_b=*/false, b,
            /*c_mod=*/(short)0, acc, /*reuse_a=*/false, /*reuse_b=*/false);
    }
    if (lane < 16) {
#pragma unroll
        for (int j = 0; j < 8; ++j)            // VGPR j == batch row j
            xz[(size_t)j * (2 * DI) + n0 + lane] = acc[j];
    }
}

// ---------------------------------------------------------------------------
// Kernel 3: the entire Mamba middle (L==1 collapses the scan; dA is dead since
// h0 == 0). Grid = 16 blocks (blocks 8..15 just zero-pad y16), 256 threads.
//   xs = silu(xp*conv_w[:,3] + conv_b)          (causal conv, single live tap)
//   x_dbl = xs @ W_x^T (64 outputs)
//   delta = softplus(x_dbl[0:32] @ W_dt^T + b_dt)
//   bc    = dot(Bm, Cm) = dot(x_dbl[32:48], x_dbl[48:64])
//   y     = (delta*bc + D_param) * xs * silu(z)
// ---------------------------------------------------------------------------
__global__ void mamba_mid_kernel(const float* __restrict__ xz,
                                 const float* __restrict__ conv_w,
                                 const float* __restrict__ conv_b,
                                 const float* __restrict__ W_x,
                                 const float* __restrict__ W_dt,
                                 const float* __restrict__ b_dt,
                                 const float* __restrict__ D_param,
                                 float* __restrict__ y16 /*16 x 1024*/) {
    int b = blockIdx.x;
    int t = threadIdx.x;
    if (b >= NB) {                      // zero padding rows for GEMM2 A-matrix
#pragma unroll
        for (int i = 0; i < 4; ++i) y16[(size_t)b * DI + t + 256 * i] = 0.0f;
        return;
    }
    __shared__ float xs[DI];
    __shared__ float sz[DI];
    __shared__ float parts[4][64];
    __shared__ float xdbl[64];
    __shared__ float bc_sh;

    const float* xzb = xz + (size_t)b * (2 * DI);
    // Phase 1: conv tap + silu, and silu(z)
#pragma unroll
    for (int i = 0; i < 4; ++i) {
        int d = t + 256 * i;
        float xc = xzb[d] * conv_w[d * 4 + 3] + conv_b[d];
        xs[d] = xc / (1.0f + expf(-xc));
        float zv = xzb[DI + d];
        sz[d] = zv / (1.0f + expf(-zv));
    }
    __syncthreads();
    // Phase 2: x_dbl[e] = sum_d xs[d]*W_x[e][d], 4 partial chunks of 256
    {
        int e = t & 63;
        int p = t >> 6;
        const float* wr = W_x + (size_t)e * DI + p * 256;
        const float* xr = xs + p * 256;
        float s = 0.0f;
#pragma unroll 8
        for (int d = 0; d < 256; ++d) s += xr[d] * wr[d];
        parts[p][e] = s;
    }
    __syncthreads();
    if (t < 64) xdbl[t] = parts[0][t] + parts[1][t] + parts[2][t] + parts[3][t];
    __syncthreads();
    // Phase 3: bc = dot(Bm, Cm)
    if (t == 0) {
        float s = 0.0f;
#pragma unroll
        for (int n = 0; n < DST; ++n) s += xdbl[DTR + n] * xdbl[DTR + DST + n];
        bc_sh = s;
    }
    __syncthreads();
    // Phase 4: delta, y
    float bc = bc_sh;
#pragma unroll
    for (int i = 0; i < 4; ++i) {
        int d = t + 256 * i;
        float acc = b_dt[d];
        const float* wd = W_dt + (size_t)d * DTR;
#pragma unroll
        for (int r = 0; r < DTR; ++r) acc += xdbl[r] * wd[r];
        // numerically stable softplus
        float delta = fmaxf(acc, 0.0f) + log1pf(expf(-fabsf(acc)));
        float y = (delta * bc + D_param[d]) * xs[d] * sz[d];
        y16[(size_t)b * DI + d] = y;
    }
}

// ---------------------------------------------------------------------------
// Kernel 4: gate = sigmoid( y16(16x1024) @ W_out^T(1024x512) ), WMMA f32.
// 32 tiles of 16 cols; 4 waves/block -> 8 blocks x 128 threads.
// ---------------------------------------------------------------------------
__global__ void gemm_out_wmma(const float* __restrict__ y16,
                              const float* __restrict__ W_out,
                              float* __restrict__ gate /*8 x 512*/) {
    int lane = threadIdx.x & 31;
    int wave = threadIdx.x >> 5;
    int n0   = (blockIdx.x * 4 + wave) * 16;
    int r    = lane & 15;
    int kh   = (lane >> 4) * 2;
    const float* arow = y16 + r * DI + kh;
    const float* brow = W_out + (size_t)(n0 + r) * DI + kh;
    v8f acc = {0.f, 0.f, 0.f, 0.f, 0.f, 0.f, 0.f, 0.f};
#pragma unroll 4
    for (int k0 = 0; k0 < DI; k0 += 4) {
        v2f a = *(const v2f*)(arow + k0);
        v2f b = *(const v2f*)(brow + k0);
        acc = __builtin_amdgcn_wmma_f32_16x16x4_f32(
            false, a, false, b, (short)0, acc, false, false);
    }
    if (lane < 16) {
#pragma unroll
        for (int j = 0; j < 8; ++j) {
            float v = acc[j];
            gate[(size_t)j * CM + n0 + lane] = 1.0f / (1.0f + expf(-v));
        }
    }
}

// ---------------------------------------------------------------------------
// Kernel 5: out = x * gate[b,c].  The bandwidth kernel: 256 MB in + 256 MB out,
// float4 coalesced, one float4 per thread, exact grid (no bounds check).
// float4 index i -> (b*512+c) = i >> 12  (16384/4 = 4096 float4 per channel).
// ---------------------------------------------------------------------------
__global__ void scale_kernel(const float* __restrict__ x,
                             const float* __restrict__ gate,
                             float* __restrict__ out) {
    size_t i = (size_t)blockIdx.x * blockDim.x + threadIdx.x;  // float4 index
    float4 v = ((const float4*)x)[i];
    float g = gate[i >> 12];
    float4 o;
    o.x = v.x * g; o.y = v.y * g; o.z = v.z * g; o.w = v.w * g;
    ((float4*)out)[i] = o;
}

// ---------------------------------------------------------------------------
// Launch.  Inputs (setup_inputs order):
//  0:x 1:W_in 2:conv_w 3:conv_b 4:W_x 5:W_dt 6:b_dt 7:A_log(dead) 8:D_param 9:W_out
// ---------------------------------------------------------------------------
extern "C" void kernel_launch(void* const* d_in, const int* in_sizes, int n_in,
                              void* d_out, int out_size, void* d_ws, size_t ws_size,
                              hipStream_t stream) {
    const float* x      = (const float*)d_in[0];
    const float* W_in   = (const float*)d_in[1];
    const float* conv_w = (const float*)d_in[2];
    const float* conv_b = (const float*)d_in[3];
    const float* W_x    = (const float*)d_in[4];
    const float* W_dt   = (const float*)d_in[5];
    const float* b_dt   = (const float*)d_in[6];
    // d_in[7] = A_log: provably unused (h0 == 0, L == 1 => dA multiplies zero)
    const float* D_par  = (const float*)d_in[8];
    const float* W_out  = (const float*)d_in[9];
    float* out = (float*)d_out;

    float* ws     = (float*)d_ws;
    float* pooled = ws;                          // 16*512   =  8192 f
    float* xz     = pooled + 16 * CM;            // 16*2048  = 32768 f
    float* y16    = xz + 16 * 2 * DI;            // 16*1024  = 16384 f
    float* gate   = y16 + 16 * DI;               //  8*512   =  4096 f

    pool_mean_kernel<<<16 * CM, 256, 0, stream>>>(x, pooled);
    gemm_xz_wmma<<<32, 128, 0, stream>>>(pooled, W_in, xz);
    mamba_mid_kernel<<<16, 256, 0, stream>>>(xz, conv_w, conv_b, W_x, W_dt,
                                             b_dt, D_par, y16);
    gemm_out_wmma<<<8, 128, 0, stream>>>(y16, W_out, gate);
    scale_kernel<<<(NB * CM * HW) / (4 * 256), 256, 0, stream>>>(x, gate, out);
}